// YOLOLoss_52243982188963
// MI455X (gfx1250) — compile-verified
//
#include <hip/hip_runtime.h>
#include <cstdint>

// ---------------------------------------------------------------------------
// YOLO loss for MI455X (gfx1250).
// Latency-bound op (12.5 MB in, scalar out). f32 precision required for the
// log/exp loss math, so WMMA is used where it is exact: V_WMMA_F32_16X16X4_F32
// (full f32 MACs) as the deterministic sum-reduction engine. Per-batch target
// metadata (2.4 KB) is staged to LDS via the gfx1250 async DMA path
// (global_load_async_to_lds_b128 + s_wait_asynccnt) since every block re-reads
// it 50x per thread.
// Workspace use: 16*640 dwords staging + 2048 float partials = 48 KB.
// ---------------------------------------------------------------------------

typedef float v2f __attribute__((ext_vector_type(2)));
typedef float v8f __attribute__((ext_vector_type(8)));

#define BS        16
#define NA        3
#define HH        104
#define WW        104
#define HW        (HH * WW)          // 10816
#define CELLS     (NA * HW)          // 32448
#define NT        50
#define STG_DW    640                // per-batch staging dwords (padded to 160 x b128)
#define REC_OFF   200                // dword offset of records inside staging region
#define NBX       127                // blocks per batch (127*256 >= 32448)
#define NPART     (NBX * BS)         // 2032
#define NPART_PAD 2048
#define WS_PART_DW (BS * STG_DW)     // 10240 dwords -> partials start here
#define EPSF      1e-7f
#define INV_BS    (1.0f / 16.0f)

// ANCHORS / stride(8). stride = 832/104 = 8, sub = 6 (in_w == 104 level).
__constant__ float AW9[9] = {1.25f, 2.0f, 4.125f, 3.75f, 7.75f, 7.375f, 14.5f, 19.5f, 46.625f};
__constant__ float AH9[9] = {1.625f, 3.75f, 2.875f, 7.625f, 5.625f, 14.875f, 11.25f, 24.75f, 40.75f};
__constant__ float AWS[3] = {14.5f, 19.5f, 46.625f};   // anchors[6..8].w / 8
__constant__ float AHS[3] = {11.25f, 24.75f, 40.75f};  // anchors[6..8].h / 8

__device__ __forceinline__ float sigmoidf_(float v) {
    return 1.0f / (1.0f + expf(-v));
}
__device__ __forceinline__ float bcef_(float p, float t) {
    p = fminf(fmaxf(p, EPSF), 1.0f - EPSF);
    return -t * logf(p) - (1.0f - t) * logf(1.0f - p);
}

// ---------------------------------------------------------------------------
// Kernel 1: build per-batch GT boxes + scatter records (compact, <=50/batch).
// Layout per batch b (dwords, at ws + b*STG_DW):
//   [0..199]   : 50 x {gx, gy, gw, gh}  (floats)
//   [200..599] : 50 x {key, tx, ty, tw, th, bls, cls, pad} (key/cls as ints)
//   [600..639] : pad (never read)
// ---------------------------------------------------------------------------
__global__ void yolo_prep_kernel(const float* __restrict__ tgt,
                                 unsigned int* __restrict__ ws) {
    int b = blockIdx.x;
    int t = threadIdx.x;
    // zero the 16 padding slots of the partial-sum array (slots 2032..2047)
    if (b == 0 && t < (NPART_PAD - NPART)) ws[WS_PART_DW + NPART + t] = 0u;
    if (t >= NT) return;

    const float* tg = tgt + (size_t)(b * NT + t) * 5;
    float t0 = tg[0], t1 = tg[1], t2 = tg[2], t3 = tg[3], t4 = tg[4];
    float gx = t0 * (float)WW, gy = t1 * (float)HH;
    float gw = t2 * (float)WW, gh = t3 * (float)HH;
    int gi = (int)floorf(gx), gj = (int)floorf(gy);

    // argmax over 9 anchors of inter/union (first max wins, like jnp.argmax)
    int best = 0; float bestv = -3.402823466e38f;
    #pragma unroll
    for (int k = 0; k < 9; ++k) {
        float inter = fminf(gw, AW9[k]) * fminf(gh, AH9[k]);
        float uni   = gw * gh + AW9[k] * AH9[k] - inter;
        float v     = inter / uni;
        if (v > bestv) { bestv = v; best = k; }
    }
    bool valid = (best >= 6) && (best < 9) && (gj < HH) && (gi < WW);
    int ln  = best - 6;
    int key = valid ? (ln * HW + gj * WW + gi) : -1;

    float tx  = gx - floorf(gx);
    float ty  = gy - floorf(gy);
    float tw  = logf(fmaxf(gw, EPSF) / AW9[best]);
    float th  = logf(fmaxf(gh, EPSF) / AH9[best]);
    float bls = 2.0f - t2 * t3;
    int   cls = (int)t4;

    unsigned int* stg = ws + (size_t)b * STG_DW;
    float* boxes = (float*)stg;
    boxes[t * 4 + 0] = gx;
    boxes[t * 4 + 1] = gy;
    boxes[t * 4 + 2] = gw;
    boxes[t * 4 + 3] = gh;
    unsigned int* rec = stg + REC_OFF + t * 8;
    rec[0] = (unsigned int)key;
    rec[1] = __float_as_uint(tx);
    rec[2] = __float_as_uint(ty);
    rec[3] = __float_as_uint(tw);
    rec[4] = __float_as_uint(th);
    rec[5] = __float_as_uint(bls);
    rec[6] = (unsigned int)cls;
    rec[7] = 0u;
}

// ---------------------------------------------------------------------------
// Kernel 2: per-cell loss over grid (NBX, BS) x 256 threads.
// ---------------------------------------------------------------------------
__global__ __launch_bounds__(256) void yolo_main_kernel(
    const float* __restrict__ in,
    const unsigned int* __restrict__ ws_stg,
    float* __restrict__ partial) {

    __shared__ __align__(16) unsigned int smem[STG_DW];
    __shared__ float red[256];

    const int b   = blockIdx.y;
    const int tid = threadIdx.x;

    // --- async DMA stage: 640 dwords = 160 x b128, 5 per lane of wave 0 ----
    if (tid < 32) {
        unsigned long long src = (unsigned long long)(uintptr_t)(ws_stg + (size_t)b * STG_DW);
        unsigned int lbase = (unsigned int)(unsigned long long)(uintptr_t)&smem[0];
        #pragma unroll
        for (int it = 0; it < 5; ++it) {
            unsigned int off  = (unsigned int)((it * 32 + tid) * 16);
            unsigned int ldsa = lbase + off;
            asm volatile("global_load_async_to_lds_b128 %0, %1, %2"
                         :: "v"(ldsa), "v"(off), "s"(src)
                         : "memory");
        }
        asm volatile("s_wait_asynccnt 0x0" ::: "memory");
    }
    __syncthreads();

    const int c = blockIdx.x * 256 + tid;
    float contrib = 0.0f;

    if (c < CELLS) {
        const int a   = c / HW;
        const int rem = c - a * HW;
        const int j   = rem / WW;
        const int i   = rem - j * WW;

        const float* pb = in + ((size_t)(b * 18 + a * 6)) * HW + rem;
        float xr = pb[0 * HW];
        float yr = pb[1 * HW];
        float wr = pb[2 * HW];
        float hr = pb[3 * HW];
        float cr = pb[4 * HW];
        float pr = pb[5 * HW];

        float x    = sigmoidf_(xr);
        float y    = sigmoidf_(yr);
        float conf = sigmoidf_(cr);
        float pcls = sigmoidf_(pr);

        float px = x + (float)i;
        float py = y + (float)j;
        float pw = expf(wr) * AWS[a];
        float ph = expf(hr) * AHS[a];

        float pax1 = px - 0.5f * pw, pax2 = px + 0.5f * pw;
        float pay1 = py - 0.5f * ph, pay2 = py + 0.5f * ph;
        float areap = pw * ph;

        const float* boxes = (const float*)smem;
        float maxiou = -3.402823466e38f;
        #pragma unroll 5
        for (int t = 0; t < NT; ++t) {
            float bx = boxes[t * 4 + 0], by = boxes[t * 4 + 1];
            float bw = boxes[t * 4 + 2], bh = boxes[t * 4 + 3];
            float ax1 = bx - 0.5f * bw, ax2 = bx + 0.5f * bw;
            float ay1 = by - 0.5f * bh, ay2 = by + 0.5f * bh;
            float iw = fmaxf(fminf(ax2, pax2) - fmaxf(ax1, pax1), 0.0f);
            float ih = fmaxf(fminf(ay2, pay2) - fmaxf(ay1, pay1), 0.0f);
            float inter = iw * ih;
            float iou = inter / (bw * bh + areap - inter);
            maxiou = fmaxf(maxiou, iou);
        }

        // last matching record wins (matches sequential scatter .set semantics)
        const unsigned int* recs = smem + REC_OFF;
        const int key = a * HW + rem;
        int hit = -1;
        for (int t = 0; t < NT; ++t)
            if ((int)recs[t * 8] == key) hit = t;

        float maskv = (hit >= 0) ? 1.0f : 0.0f;
        float noobj = (hit < 0 && maxiou <= 0.5f) ? 1.0f : 0.0f;

        contrib = bcef_(conf, maskv) * (maskv + noobj) * INV_BS;
        if (hit >= 0) {
            const unsigned int* r = recs + hit * 8;
            float tx  = __uint_as_float(r[1]);
            float ty  = __uint_as_float(r[2]);
            float tw  = __uint_as_float(r[3]);
            float th  = __uint_as_float(r[4]);
            float bls = __uint_as_float(r[5]);
            float tcls = ((int)r[6] == 0) ? 1.0f : 0.0f;
            contrib += (bcef_(x, tx) + bcef_(y, ty)) * bls * INV_BS;
            contrib += 0.5f * ((wr - tw) * (wr - tw) + (hr - th) * (hr - th)) * bls * INV_BS;
            contrib += bcef_(pcls, tcls) * INV_BS;
        }
    }

    red[tid] = contrib;
    __syncthreads();

    // --- block reduction on wave 0 via full-precision f32 WMMA -------------
    // A = ones(16x4), so D[m][n] = colsum(B)[n] (+C). Sum of all D = 16 * sum(B).
    if (tid < 32) {   // wave 0: EXEC all-ones at the WMMA (ISA requirement)
        v2f A; A.x = 1.0f; A.y = 1.0f;
        v8f C = {0.0f, 0.0f, 0.0f, 0.0f, 0.0f, 0.0f, 0.0f, 0.0f};
        #pragma unroll
        for (int q = 0; q < 4; ++q) {
            v2f Bv; Bv.x = red[q * 64 + tid]; Bv.y = red[q * 64 + 32 + tid];
            C = __builtin_amdgcn_wmma_f32_16x16x4_f32(false, A, false, Bv,
                                                      (short)0, C, false, false);
        }
        float s = C[0] + C[1] + C[2] + C[3] + C[4] + C[5] + C[6] + C[7];
        #pragma unroll
        for (int o = 16; o > 0; o >>= 1) s += __shfl_xor(s, o, 32);
        if (tid == 0)
            partial[blockIdx.y * NBX + blockIdx.x] = s * (1.0f / 16.0f);
    }
}

// ---------------------------------------------------------------------------
// Kernel 3: deterministic final reduction of 2048 (padded) partials, 1 wave.
// ---------------------------------------------------------------------------
__global__ void yolo_final_kernel(const float* __restrict__ partial,
                                  float* __restrict__ out) {
    const int l = threadIdx.x;  // 32 threads, full wave
    v2f A; A.x = 1.0f; A.y = 1.0f;
    v8f C = {0.0f, 0.0f, 0.0f, 0.0f, 0.0f, 0.0f, 0.0f, 0.0f};
    for (int q = 0; q < NPART_PAD / 64; ++q) {
        v2f Bv; Bv.x = partial[q * 64 + l]; Bv.y = partial[q * 64 + 32 + l];
        C = __builtin_amdgcn_wmma_f32_16x16x4_f32(false, A, false, Bv,
                                                  (short)0, C, false, false);
    }
    float s = C[0] + C[1] + C[2] + C[3] + C[4] + C[5] + C[6] + C[7];
    #pragma unroll
    for (int o = 16; o > 0; o >>= 1) s += __shfl_xor(s, o, 32);
    if (l == 0) out[0] = s * (1.0f / 16.0f);
}

// ---------------------------------------------------------------------------
extern "C" void kernel_launch(void* const* d_in, const int* in_sizes, int n_in,
                              void* d_out, int out_size, void* d_ws, size_t ws_size,
                              hipStream_t stream) {
    (void)in_sizes; (void)n_in; (void)out_size; (void)ws_size;
    const float* in  = (const float*)d_in[0];   // (16, 18, 104, 104) f32
    const float* tgt = (const float*)d_in[1];   // (16, 50, 5) f32
    unsigned int* ws = (unsigned int*)d_ws;     // needs >= 48 KB
    float* partial   = (float*)(ws + WS_PART_DW);

    yolo_prep_kernel<<<BS, 64, 0, stream>>>(tgt, ws);
    dim3 grid(NBX, BS);
    yolo_main_kernel<<<grid, 256, 0, stream>>>(in, ws, partial);
    yolo_final_kernel<<<1, 32, 0, stream>>>(partial, (float*)d_out);
}